// intopt_knapsack_solver_61658550502069
// MI455X (gfx1250) — compile-verified
//
#include <hip/hip_runtime.h>

// ---------------------------------------------------------------------------
// Batched log-barrier IPM knapsack solver for MI455X (gfx1250, wave32).
// One workgroup (256 thr = 8 wave32) per batch row; all per-row state
// (c, w, x : 16 floats/thread) lives in VGPRs for all 41 iterations.
// Cross-wave 3-way sum reduction via V_WMMA_F32_16X16X4_F32 (partials in
// A rows 0..5, B = ones -> row sums in D); intra-wave reductions via
// ds_swizzle XOR butterflies; 2 v_rcp_f32 per element per iteration using
// p = x(1-x) algebra; fully branch-free inner loops.
// ---------------------------------------------------------------------------

#define N_ITEMS  4096
#define TPB      256
#define EPT      16            // elements per thread = N_ITEMS / TPB
#define NWAVE    (TPB / 32)    // 8 wave32 per workgroup
#define CAPACITY 600.0f
#define DAMPING  1.0e-3f
#define N_ITER   40

typedef __attribute__((ext_vector_type(2))) float v2f;
typedef __attribute__((ext_vector_type(8))) float v8f;

__device__ __forceinline__ float frcp(float x) { return __builtin_amdgcn_rcpf(x); }

__device__ __forceinline__ float bcast_lane0(float v) {
  return __int_as_float(__builtin_amdgcn_readfirstlane(__float_as_int(v)));
}

// ds_swizzle imm (group-of-32 mode): and=0x1F, or=0, xor=mask<<10
#define SWZ(v, imm) __int_as_float(__builtin_amdgcn_ds_swizzle(__float_as_int(v), (imm)))

__device__ __forceinline__ float wave_sum32(float v) {
  v += SWZ(v, 0x041F);  // xor 1
  v += SWZ(v, 0x081F);  // xor 2
  v += SWZ(v, 0x101F);  // xor 4
  v += SWZ(v, 0x201F);  // xor 8
  v += SWZ(v, 0x401F);  // xor 16
  return v;
}

__device__ __forceinline__ float wave_max32(float v) {
  v = fmaxf(v, SWZ(v, 0x041F));
  v = fmaxf(v, SWZ(v, 0x081F));
  v = fmaxf(v, SWZ(v, 0x101F));
  v = fmaxf(v, SWZ(v, 0x201F));
  v = fmaxf(v, SWZ(v, 0x401F));
  return v;
}

// Reduce NWAVE(=8) LDS floats: 1 ds_load + 3-step XOR butterfly (all lanes
// end up with the result; lanes 0-7,8-15,... each cover indices 0..7).
__device__ __forceinline__ float lds8_sum(const float* p, int lane) {
  float v = p[lane & 7];
  v += SWZ(v, 0x041F);
  v += SWZ(v, 0x081F);
  v += SWZ(v, 0x101F);
  return v;
}
__device__ __forceinline__ float lds8_max(const float* p, int lane) {
  float v = p[lane & 7];
  v = fmaxf(v, SWZ(v, 0x041F));
  v = fmaxf(v, SWZ(v, 0x081F));
  v = fmaxf(v, SWZ(v, 0x101F));
  return v;
}

// Cross-wave reduction of three sums via one V_WMMA_F32_16X16X4_F32.
// pS[q*NWAVE + w] holds per-wave partial of quantity q (q=0..2, w=0..7).
// A layout (ISA 7.12.2, 32-bit A 16x4): lanes 0-15 -> M=lane, K={0,1};
// lanes 16-31 -> M=lane-16, K={2,3}.  Row 2q+hh carries p_q[4*hh .. 4*hh+3].
// B = ones(4x16) -> D[m][*] = row-sum of A.  D layout: VGPR j, lanes 0-15
// hold M=j, so lane 0 has all six row sums in d[0..5].  Branch-free A build.
__device__ __forceinline__ void xwave_sum3_wmma(const float* pS,
                                                float& S0, float& S1, float& S2) {
  const int lane  = threadIdx.x & 31;
  const int half  = lane >> 4;   // 0: K=0,1   1: K=2,3
  const int r     = lane & 15;   // A row
  const bool valid = (r < 6);
  const int q  = valid ? (r >> 1) : 0;   // clamp so address is always legal
  const int j0 = 4 * (r & 1) + 2 * half;
  const float ax = pS[q * NWAVE + j0];
  const float ay = pS[q * NWAVE + j0 + 1];
  v2f a;
  a.x = valid ? ax : 0.0f;
  a.y = valid ? ay : 0.0f;
  v2f b; b.x = 1.0f; b.y = 1.0f;
  v8f cz = {};
  // 8 args: (neg_a, A, neg_b, B, c_mod, C, reuse_a, reuse_b)
  v8f d = __builtin_amdgcn_wmma_f32_16x16x4_f32(
      false, a, false, b, (short)0, cz, false, false);
  S0 = bcast_lane0(d[0] + d[1]);
  S1 = bcast_lane0(d[2] + d[3]);
  S2 = bcast_lane0(d[4] + d[5]);
}

__global__ __launch_bounds__(TPB) void ipm_knapsack_kernel(
    const float* __restrict__ costs,
    const float* __restrict__ weights,
    float* __restrict__ out) {
  __shared__ float pS[3 * NWAVE];
  __shared__ float pT[NWAVE];

  const int t    = threadIdx.x;
  const int wv   = t >> 5;
  const int lane = t & 31;
  const size_t row = (size_t)blockIdx.x * N_ITEMS;

  float c[EPT], w[EPT], x[EPT];

  // One coalesced pass over the cost row (float4 per thread, x4 chunks).
  const float4* c4 = (const float4*)(costs + row);
  const float4* w4 = (const float4*)weights;
  float wpart = 0.0f;
#pragma unroll
  for (int i = 0; i < 4; ++i) {
    float4 cv = c4[i * TPB + t];
    float4 ww = w4[i * TPB + t];
    c[4 * i + 0] = -cv.x; c[4 * i + 1] = -cv.y;
    c[4 * i + 2] = -cv.z; c[4 * i + 3] = -cv.w;
    w[4 * i + 0] = ww.x;  w[4 * i + 1] = ww.y;
    w[4 * i + 2] = ww.z;  w[4 * i + 3] = ww.w;
    wpart += (ww.x + ww.y) + (ww.z + ww.w);
  }

  // x0 = capacity / sum(weights)
  float wsum = wave_sum32(wpart);
  if (lane == 0) pT[wv] = wsum;
  __syncthreads();
  const float wall = lds8_sum(pT, lane);
  const float x0   = CAPACITY * frcp(wall);
#pragma unroll
  for (int e = 0; e < EPT; ++e) x[e] = x0;

  float lam = 0.0f;
  float mu  = 1.0f;

  for (int it = 0; it < N_ITER; ++it) {
    float F1a[EPT], iHa[EPT], rpa[EPT];
    float s1 = 0.0f, s2 = 0.0f, sxw = 0.0f;
#pragma unroll
    for (int e = 0; e < EPT; ++e) {
      const float xe  = x[e];
      const float we  = w[e];
      const float p   = fmaf(-xe, xe, xe);            // x(1-x)
      const float rp  = frcp(p);                      // 1/(x(1-x))
      const float g   = fmaf(2.0f, xe, -1.0f) * rp;   // 1/(1-x) - 1/x
      const float F1  = fmaf(mu, g, fmaf(lam, we, c[e]));
      const float q2  = fmaf(-2.0f, p, 1.0f);         // x^2 + (1-x)^2
      const float iH  = (p * p) * frcp(mu * q2);      // 1/H, H = mu*q2/p^2
      const float wiH = we * iH;
      s1  = fmaf(wiH, F1, s1);
      s2  = fmaf(wiH, we, s2);
      sxw = fmaf(xe, we, sxw);
      F1a[e] = F1;
      iHa[e] = iH;
      rpa[e] = rp;
    }
    s1  = wave_sum32(s1);
    s2  = wave_sum32(s2);
    sxw = wave_sum32(sxw);
    if (lane == 0) {
      pS[0 * NWAVE + wv] = s1;
      pS[1 * NWAVE + wv] = s2;
      pS[2 * NWAVE + wv] = sxw;
    }
    __syncthreads();
    float S1, S2, XW;
    xwave_sum3_wmma(pS, S1, S2, XW);
    const float F2   = XW - CAPACITY;
    const float dlam = (F2 - S1) * frcp(S2);

    // Fraction-to-boundary: min_e sel/|dx| == 1 / max_e (|dx| * osel * rp),
    // osel = dx>0 ? x : 1-x  (since 1/(1-x)=x*rp, 1/x=(1-x)*rp). Branch-free.
    float dxa[EPT];
    float tmax = 0.0f;
#pragma unroll
    for (int e = 0; e < EPT; ++e) {
      const float dx   = -fmaf(dlam, w[e], F1a[e]) * iHa[e];
      dxa[e] = dx;
      const float osel  = (dx > 0.0f) ? x[e] : (1.0f - x[e]);
      const float ratio = fabsf(dx) * osel * rpa[e];
      tmax = fmaxf(tmax, ratio);
    }
    tmax = wave_max32(tmax);
    if (lane == 0) pT[wv] = tmax;
    __syncthreads();
    const float tall  = lds8_max(pT, lane);
    const float alpha = fminf(1.0f, 0.99f * frcp(tall));
#pragma unroll
    for (int e = 0; e < EPT; ++e) x[e] = fmaf(alpha, dxa[e], x[e]);
    lam = fmaf(alpha, dlam, lam);
    mu  = fmaxf(mu * 0.5f, DAMPING);
  }

  // One damped-KKT refine step at mu = DAMPING; out = x + dx.
  {
    float F1a[EPT], iHa[EPT];
    float s1 = 0.0f, s2 = 0.0f, sxw = 0.0f;
#pragma unroll
    for (int e = 0; e < EPT; ++e) {
      const float xe  = x[e];
      const float we  = w[e];
      const float p   = fmaf(-xe, xe, xe);
      const float rp  = frcp(p);
      const float g   = fmaf(2.0f, xe, -1.0f) * rp;
      const float F1  = fmaf(DAMPING, g, fmaf(lam, we, c[e]));
      const float q2  = fmaf(-2.0f, p, 1.0f);
      const float iH  = (p * p) * frcp(DAMPING * q2);
      const float wiH = we * iH;
      s1  = fmaf(wiH, F1, s1);
      s2  = fmaf(wiH, we, s2);
      sxw = fmaf(xe, we, sxw);
      F1a[e] = F1;
      iHa[e] = iH;
    }
    s1  = wave_sum32(s1);
    s2  = wave_sum32(s2);
    sxw = wave_sum32(sxw);
    if (lane == 0) {
      pS[0 * NWAVE + wv] = s1;
      pS[1 * NWAVE + wv] = s2;
      pS[2 * NWAVE + wv] = sxw;
    }
    __syncthreads();
    float S1, S2, XW;
    xwave_sum3_wmma(pS, S1, S2, XW);
    const float F2   = XW - CAPACITY;
    const float dlam = (F2 - S1) * frcp(S2);

    float4* o4 = (float4*)(out + row);
#pragma unroll
    for (int i = 0; i < 4; ++i) {
      float4 ov;
      ov.x = fmaf(-fmaf(dlam, w[4 * i + 0], F1a[4 * i + 0]), iHa[4 * i + 0], x[4 * i + 0]);
      ov.y = fmaf(-fmaf(dlam, w[4 * i + 1], F1a[4 * i + 1]), iHa[4 * i + 1], x[4 * i + 1]);
      ov.z = fmaf(-fmaf(dlam, w[4 * i + 2], F1a[4 * i + 2]), iHa[4 * i + 2], x[4 * i + 2]);
      ov.w = fmaf(-fmaf(dlam, w[4 * i + 3], F1a[4 * i + 3]), iHa[4 * i + 3], x[4 * i + 3]);
      o4[i * TPB + t] = ov;
    }
  }
}

extern "C" void kernel_launch(void* const* d_in, const int* in_sizes, int n_in,
                              void* d_out, int out_size, void* d_ws, size_t ws_size,
                              hipStream_t stream) {
  const float* costs   = (const float*)d_in[0];   // [B, 4096] fp32
  const float* weights = (const float*)d_in[1];   // [4096] fp32
  float* out = (float*)d_out;                     // [B, 4096] fp32
  (void)d_ws; (void)ws_size; (void)n_in; (void)out_size;

  const int n = in_sizes[1];            // 4096
  const int B = in_sizes[0] / n;        // 2048

  ipm_knapsack_kernel<<<dim3(B), dim3(TPB), 0, stream>>>(costs, weights, out);
}